// EPA_15427522527870
// MI455X (gfx1250) — compile-verified
//
#include <hip/hip_runtime.h>

typedef float v2f __attribute__((ext_vector_type(2)));
typedef float v8f __attribute__((ext_vector_type(8)));

#define WMMA_F32(a, b, c) \
  __builtin_amdgcn_wmma_f32_16x16x4_f32(false, (a), false, (b), (short)0, (c), false, false)

// Problem constants
#define BATCH 4
#define NTOK 4096
#define CCH 512
#define HEADS 8
#define DH 64        // C / H
#define KL 64        // Linformer K
#define MROWS (BATCH * NTOK)   // 16384
#define OCH (4 * CCH)          // 2048
#define BH (BATCH * HEADS)     // 32

// Workspace offsets (in floats)
#define OFF_Q   ((size_t)0)
#define OFF_K   ((size_t)8388608)
#define OFF_VCA ((size_t)16777216)
#define OFF_VSA ((size_t)25165824)
#define OFF_KP  ((size_t)33554432)
#define OFF_VP  ((size_t)33685504)
#define OFF_P   ((size_t)33816576)

// ---------------------------------------------------------------------------
// K1: qkvv = x @ W^T, scattered into Q/K/Vca/Vsa as [B,H,d,N]
// grid (128, 32), block 256 (8 waves). Block tile: 128 M x 64 O.
// Wave w -> M sub-tile w, 4 O sub-tiles.
// ---------------------------------------------------------------------------
__global__ __launch_bounds__(256) void qkvv_gemm(const float* __restrict__ x,
                                                 const float* __restrict__ W,
                                                 float* __restrict__ ws) {
  const int lane = threadIdx.x & 31;
  const int wave = threadIdx.x >> 5;
  const int l16  = lane & 15;
  const int hi   = lane >> 4;            // 0: K=0,1   1: K=2,3
  const int mBase = blockIdx.x * 128 + wave * 16;
  const int oBase = blockIdx.y * 64;

  const float* xRow = x + (size_t)(mBase + l16) * CCH + (hi ? 2 : 0);
  v8f acc[4] = {};
  for (int c0 = 0; c0 < CCH; c0 += 4) {
    v2f a = *(const v2f*)(xRow + c0);
#pragma unroll
    for (int t = 0; t < 4; ++t) {
      const int o = oBase + t * 16 + l16;
      v2f b = *(const v2f*)(W + (size_t)o * CCH + c0 + (hi ? 2 : 0));
      acc[t] = WMMA_F32(a, b, acc[t]);
    }
  }
#pragma unroll
  for (int t = 0; t < 4; ++t) {
    const int o  = oBase + t * 16 + l16;
    const int p  = o >> 9;
    const int h  = (o >> 6) & 7;
    const int dd = o & 63;
#pragma unroll
    for (int v = 0; v < 8; ++v) {
      const int m  = mBase + v + (hi ? 8 : 0);
      const int b_ = m >> 12;
      const int n  = m & (NTOK - 1);
      const size_t dst = (size_t)p * 8388608 +
                         ((size_t)(b_ * HEADS + h) * DH + dd) * NTOK + n;
      ws[dst] = acc[t][v];
    }
  }
}

// ---------------------------------------------------------------------------
// K2: k_proj = Kt @ EF, v_proj = Vsa @ EF   ([d,N] x [N,K] -> [d,K] per bh)
// grid 64 (= 32 bh * 2 which), block 512 (16 waves), 1 tile / wave.
// ---------------------------------------------------------------------------
__global__ __launch_bounds__(512) void ef_proj(float* __restrict__ ws,
                                               const float* __restrict__ EF) {
  const int lane = threadIdx.x & 31;
  const int wave = threadIdx.x >> 5;
  const int l16  = lane & 15;
  const int hi   = lane >> 4;
  const int which = blockIdx.x & 1;
  const int bh    = blockIdx.x >> 1;
  const int dt = wave >> 2, kt = wave & 3;

  const float* src = ws + (which ? OFF_VSA : OFF_K) + (size_t)bh * DH * NTOK;
  float* dst       = ws + (which ? OFF_VP : OFF_KP) + (size_t)bh * DH * KL;

  const float* aRow = src + (size_t)(dt * 16 + l16) * NTOK + (hi ? 2 : 0);
  const int kcol = kt * 16 + l16;
  v8f acc = {};
  for (int n0 = 0; n0 < NTOK; n0 += 4) {
    v2f a = *(const v2f*)(aRow + n0);
    const int k0 = n0 + (hi ? 2 : 0);
    v2f b;
    b.x = EF[(size_t)k0 * KL + kcol];
    b.y = EF[(size_t)(k0 + 1) * KL + kcol];
    acc = WMMA_F32(a, b, acc);
  }
#pragma unroll
  for (int v = 0; v < 8; ++v)
    dst[(size_t)(dt * 16 + v + (hi ? 8 : 0)) * KL + kcol] = acc[v];
}

// ---------------------------------------------------------------------------
// K3: in-place L2 normalization of Q and K rows over N (len 4096)
// grid 4096 (2048 Q rows + 2048 K rows), block 256.
// ---------------------------------------------------------------------------
__global__ __launch_bounds__(256) void l2norm_rows(float* __restrict__ ws) {
  __shared__ float red[256];
  const int tid = threadIdx.x;
  const int row = blockIdx.x;
  float* ptr = ws + (row < 2048 ? OFF_Q : OFF_K) +
               (size_t)(row & 2047) * NTOK;
  float s = 0.f;
#pragma unroll
  for (int i = 0; i < 16; ++i) {
    float v = ptr[tid + i * 256];
    s += v * v;
  }
  red[tid] = s;
  __syncthreads();
  for (int off = 128; off > 0; off >>= 1) {
    if (tid < off) red[tid] += red[tid + off];
    __syncthreads();
  }
  const float scale = 1.0f / fmaxf(sqrtf(red[0]), 1e-12f);
#pragma unroll
  for (int i = 0; i < 16; ++i) ptr[tid + i * 256] *= scale;
}

// ---------------------------------------------------------------------------
// K4: S = qn @ kn^T (contraction over N), * temperature, row softmax -> Pca
// grid 32 (bh), block 512 (16 waves, one 16x16 tile each).
// ---------------------------------------------------------------------------
__global__ __launch_bounds__(512) void ca_scores(float* __restrict__ ws,
                                                 const float* __restrict__ temperature) {
  __shared__ float S[64][64];
  const int lane = threadIdx.x & 31;
  const int wave = threadIdx.x >> 5;
  const int l16  = lane & 15;
  const int hi   = lane >> 4;
  const int bh = blockIdx.x;
  const int h  = bh & 7;
  const int dt = wave >> 2, et = wave & 3;

  const float* qRow = ws + OFF_Q + (size_t)bh * DH * NTOK +
                      (size_t)(dt * 16 + l16) * NTOK + (hi ? 2 : 0);
  const float* kRow = ws + OFF_K + (size_t)bh * DH * NTOK +
                      (size_t)(et * 16 + l16) * NTOK + (hi ? 2 : 0);
  v8f acc = {};
  for (int n0 = 0; n0 < NTOK; n0 += 4) {
    v2f a = *(const v2f*)(qRow + n0);
    v2f b = *(const v2f*)(kRow + n0);
    acc = WMMA_F32(a, b, acc);
  }
  const float tsc = temperature[h];
#pragma unroll
  for (int v = 0; v < 8; ++v)
    S[dt * 16 + v + (hi ? 8 : 0)][et * 16 + l16] = acc[v] * tsc;
  __syncthreads();

  if (threadIdx.x < 64) {
    const int r = threadIdx.x;
    float m = S[r][0];
#pragma unroll
    for (int j = 1; j < 64; ++j) m = fmaxf(m, S[r][j]);
    float sum = 0.f;
    float e[64];
#pragma unroll
    for (int j = 0; j < 64; ++j) { e[j] = __expf(S[r][j] - m); sum += e[j]; }
    const float inv = 1.0f / sum;
    float* dst = ws + OFF_P + (size_t)bh * DH * DH + (size_t)r * DH;
#pragma unroll
    for (int j = 0; j < 64; ++j) dst[j] = e[j] * inv;
  }
}

// ---------------------------------------------------------------------------
// K5: x_CA = P @ Vca, written to out[b][n][h*64+d]  (initializes all of out)
// grid (32 nchunks, 32 bh), block 256 (8 waves, 4 tiles each).
// ---------------------------------------------------------------------------
__global__ __launch_bounds__(256) void ca_out(const float* __restrict__ ws,
                                              float* __restrict__ out) {
  __shared__ float P[64][64];
  const int tid  = threadIdx.x;
  const int lane = tid & 31;
  const int wave = tid >> 5;
  const int l16  = lane & 15;
  const int hi   = lane >> 4;
  const int bh = blockIdx.y;
  const int b_ = bh >> 3;
  const int h  = bh & 7;
  const int nBase = blockIdx.x * 128;

  const float* Pg = ws + OFF_P + (size_t)bh * DH * DH;
  for (int i = tid; i < 4096; i += 256) P[i >> 6][i & 63] = Pg[i];
  __syncthreads();

  const int dt = wave >> 1;
  const int ng = wave & 1;
  const int mRow = dt * 16 + l16;
  const float* vca = ws + OFF_VCA + (size_t)bh * DH * NTOK;

  v8f acc[4] = {};
  for (int e0 = 0; e0 < DH; e0 += 4) {
    const int k0 = e0 + (hi ? 2 : 0);
    v2f a;
    a.x = P[mRow][k0];
    a.y = P[mRow][k0 + 1];
#pragma unroll
    for (int t = 0; t < 4; ++t) {
      const int n = nBase + (ng * 4 + t) * 16 + l16;
      v2f bfrag;
      bfrag.x = vca[(size_t)k0 * NTOK + n];
      bfrag.y = vca[(size_t)(k0 + 1) * NTOK + n];
      acc[t] = WMMA_F32(a, bfrag, acc[t]);
    }
  }
#pragma unroll
  for (int t = 0; t < 4; ++t) {
    const int n = nBase + (ng * 4 + t) * 16 + l16;
#pragma unroll
    for (int v = 0; v < 8; ++v) {
      const int d = dt * 16 + v + (hi ? 8 : 0);
      out[((size_t)(b_ * NTOK + n)) * CCH + h * DH + d] = acc[t][v];
    }
  }
}

// ---------------------------------------------------------------------------
// K6: SA: L = qn^T @ k_proj, *temp2, row softmax, then @ v_proj^T,
// scatter-added into out via the faithful permute(0,3,1,2)+reshape mapping.
// grid (32 nchunks, 32 bh), block 256 (8 waves).
// ---------------------------------------------------------------------------
__global__ __launch_bounds__(256) void sa_attn(float* __restrict__ ws,
                                               const float* __restrict__ temperature2,
                                               float* __restrict__ out) {
  __shared__ float P[128][64];
  const int tid  = threadIdx.x;
  const int lane = tid & 31;
  const int wave = tid >> 5;
  const int l16  = lane & 15;
  const int hi   = lane >> 4;
  const int bh = blockIdx.y;
  const int b_ = bh >> 3;
  const int h  = bh & 7;
  const int nBase = blockIdx.x * 128;

  const float* Q  = ws + OFF_Q  + (size_t)bh * DH * NTOK;
  const float* Kp = ws + OFF_KP + (size_t)bh * DH * KL;
  const float* Vp = ws + OFF_VP + (size_t)bh * DH * KL;
  const float t2 = temperature2[h];

  // Phase 1: logits [128, 64]
  {
    const int nGlob = nBase + wave * 16 + l16;   // A row (token)
    v8f acc[4] = {};
    for (int d0 = 0; d0 < DH; d0 += 4) {
      const int k0 = d0 + (hi ? 2 : 0);
      v2f a;
      a.x = Q[(size_t)k0 * NTOK + nGlob];
      a.y = Q[(size_t)(k0 + 1) * NTOK + nGlob];
#pragma unroll
      for (int t = 0; t < 4; ++t) {
        const int kc = t * 16 + l16;
        v2f bfrag;
        bfrag.x = Kp[(size_t)k0 * KL + kc];
        bfrag.y = Kp[(size_t)(k0 + 1) * KL + kc];
        acc[t] = WMMA_F32(a, bfrag, acc[t]);
      }
    }
#pragma unroll
    for (int t = 0; t < 4; ++t)
#pragma unroll
      for (int v = 0; v < 8; ++v)
        P[wave * 16 + v + (hi ? 8 : 0)][t * 16 + l16] = acc[t][v] * t2;
  }
  __syncthreads();

  // softmax over K (64) per row
  if (tid < 128) {
    float m = P[tid][0];
#pragma unroll
    for (int j = 1; j < 64; ++j) m = fmaxf(m, P[tid][j]);
    float sum = 0.f;
#pragma unroll
    for (int j = 0; j < 64; ++j) { float e = __expf(P[tid][j] - m); P[tid][j] = e; sum += e; }
    const float inv = 1.0f / sum;
#pragma unroll
    for (int j = 0; j < 64; ++j) P[tid][j] *= inv;
  }
  __syncthreads();

  // Phase 2: x_SA = P @ Vp^T, scatter-add
  {
    const int mRow = wave * 16 + l16;
    v8f acc[4] = {};
    for (int kk0 = 0; kk0 < KL; kk0 += 4) {
      const int k0 = kk0 + (hi ? 2 : 0);
      v2f a;
      a.x = P[mRow][k0];
      a.y = P[mRow][k0 + 1];
#pragma unroll
      for (int t = 0; t < 4; ++t) {
        const int dd = t * 16 + l16;
        v2f bfrag = *(const v2f*)(Vp + (size_t)dd * KL + k0);
        acc[t] = WMMA_F32(a, bfrag, acc[t]);
      }
    }
#pragma unroll
    for (int t = 0; t < 4; ++t) {
      const int dd = t * 16 + l16;
#pragma unroll
      for (int v = 0; v < 8; ++v) {
        const int n = nBase + wave * 16 + v + (hi ? 8 : 0);
        const size_t f = (size_t)dd * (HEADS * NTOK) + (size_t)h * NTOK + n;
        out[(size_t)b_ * (NTOK * CCH) + f] += acc[t][v];
      }
    }
  }
}

// ---------------------------------------------------------------------------
extern "C" void kernel_launch(void* const* d_in, const int* in_sizes, int n_in,
                              void* d_out, int out_size, void* d_ws, size_t ws_size,
                              hipStream_t stream) {
  const float* x  = (const float*)d_in[0];
  const float* W  = (const float*)d_in[1];
  const float* EF = (const float*)d_in[2];
  const float* t1 = (const float*)d_in[3];
  const float* t2 = (const float*)d_in[4];
  float* out = (float*)d_out;
  float* ws  = (float*)d_ws;

  qkvv_gemm  <<<dim3(128, 32), 256, 0, stream>>>(x, W, ws);
  ef_proj    <<<64,            512, 0, stream>>>(ws, EF);
  l2norm_rows<<<4096,          256, 0, stream>>>(ws);
  ca_scores  <<<32,            512, 0, stream>>>(ws, t1);
  ca_out     <<<dim3(32, 32),  256, 0, stream>>>(ws, out);
  sa_attn    <<<dim3(32, 32),  256, 0, stream>>>(ws, t2, out);
}